// TokenSelector_26096221291058
// MI455X (gfx1250) — compile-verified
//
#include <hip/hip_runtime.h>
#include <stdint.h>

#define B_   32
#define D_   256
#define H_   64
#define W_   64
#define N_   (H_ * W_)   // 4096 tokens per batch
#define K_   (N_ / 2)    // 2048 kept
#define LN_EPS 1e-5f

typedef __attribute__((ext_vector_type(2))) float v2f;
typedef __attribute__((ext_vector_type(8))) float v8f;

// ---------------------------------------------------------------------------
// Kernel 1: per-token reductions via V_WMMA_F32_16X16X4_F32.
//   For each token n: S1 = sum_d t, Sw = sum_d t*wf, S2 = sum_d t^2,
//   score ~ rsqrt(S2/D - mu^2 + eps) * (Sw - mu*C1)   (ordering-equivalent)
//   Emit order-preserving uint32 key.
// grid = (N/128, B), block = 256 (8 waves, 16 tokens per wave)
// ---------------------------------------------------------------------------
__global__ __launch_bounds__(256)
void ts_score_kernel(const float* __restrict__ tokens,
                     const float* __restrict__ norm_w,
                     const float* __restrict__ fc_w,
                     uint32_t* __restrict__ uscore)
{
    __shared__ float s_wf[D_];
    __shared__ float s_c1;
    const int tid = threadIdx.x;

    s_wf[tid] = norm_w[tid] * fc_w[tid];
    __syncthreads();
    if (tid == 0) {
        float c = 0.f;
        for (int d = 0; d < D_; ++d) c += s_wf[d];
        s_c1 = c;
    }
    __syncthreads();
    const float C1 = s_c1;

    const int lane = tid & 31;
    const int wave = tid >> 5;
    const int b    = blockIdx.y;
    const int n0   = blockIdx.x * 128 + wave * 16;

    // A-matrix 16x4 layout: lane -> row M = lane&15, K-pair = (lane<16)?(0,1):(2,3)
    const int m  = lane & 15;
    const int kb = (lane < 16) ? 0 : 2;
    // B-matrix 4x16 layout: lane -> col N = lane&15, K-pair = (lane<16)?(0,1):(2,3)
    const int j  = lane & 15;

    const float* __restrict__ tb = tokens + (size_t)b * D_ * N_;

    v8f acc1 = {0.f, 0.f, 0.f, 0.f, 0.f, 0.f, 0.f, 0.f};
    v8f acc2 = {0.f, 0.f, 0.f, 0.f, 0.f, 0.f, 0.f, 0.f};

    v2f a2;                       // constant A for the S2 accumulator: row0 = ones
    a2.x = (m == 0) ? 1.f : 0.f;
    a2.y = a2.x;

    for (int k0 = 0; k0 < D_; k0 += 4) {
        const float wf0 = s_wf[k0 + kb];       // LDS broadcast per half-wave
        const float wf1 = s_wf[k0 + kb + 1];

        v2f a1;                                 // row0 = ones (S1), row1 = wf (Sw)
        a1.x = (m == 0) ? 1.f : ((m == 1) ? wf0 : 0.f);
        a1.y = (m == 0) ? 1.f : ((m == 1) ? wf1 : 0.f);

        const float* p = tb + (size_t)(k0 + kb) * N_ + n0 + j;
        v2f bm;
        bm.x = p[0];                            // T[k0+kb  ][n0+j]
        bm.y = p[N_];                           // T[k0+kb+1][n0+j]
        v2f b2;
        b2.x = bm.x * bm.x;
        b2.y = bm.y * bm.y;

        acc1 = __builtin_amdgcn_wmma_f32_16x16x4_f32(false, a1, false, bm,
                                                     (short)0, acc1, false, false);
        acc2 = __builtin_amdgcn_wmma_f32_16x16x4_f32(false, a2, false, b2,
                                                     (short)0, acc2, false, false);
    }

    // C/D layout: lane l<16, VGPR r -> D[M=r][N=l]. So lane l owns token n0+l:
    if (lane < 16) {
        const float S1 = acc1[0];
        const float Sw = acc1[1];
        const float S2 = acc2[0];
        const float mu  = S1 * (1.0f / D_);
        float var = S2 * (1.0f / D_) - mu * mu;
        var = fmaxf(var, 0.0f);
        const float rs = rsqrtf(var + LN_EPS);
        const float score = rs * (Sw - mu * C1);
        uint32_t u = __float_as_uint(score);
        u ^= (u >> 31) ? 0xFFFFFFFFu : 0x80000000u;   // order-preserving map
        uscore[(size_t)b * N_ + n0 + lane] = u;
    }
}

// ---------------------------------------------------------------------------
// Block-wide exclusive scan helper: 256 threads, 16 contiguous values each.
// Returns the total. spart is a shared int[256].
// ---------------------------------------------------------------------------
__device__ __forceinline__ int ts_block_exscan16(int* vals, int* spart, int tid)
{
    int run = 0;
    for (int i = 0; i < 16; ++i) { int t = vals[i]; vals[i] = run; run += t; }
    spart[tid] = run;
    __syncthreads();
    for (int off = 1; off < 256; off <<= 1) {
        int v = (tid >= off) ? spart[tid - off] : 0;
        __syncthreads();
        spart[tid] += v;
        __syncthreads();
    }
    const int base  = spart[tid] - run;   // exclusive base for this thread
    const int total = spart[255];
    for (int i = 0; i < 16; ++i) vals[i] += base;
    __syncthreads();                       // spart reusable after return
    return total;
}

// ---------------------------------------------------------------------------
// Kernel 2: exact top-K per batch via 4x8-bit radix select + stable compaction.
// Reads u-keys, writes keeppos[b][n] (position 0..K-1 in ascending-index order,
// or -1) IN PLACE over the same buffer. grid = B, block = 256.
// ---------------------------------------------------------------------------
__global__ __launch_bounds__(256)
void ts_topk_kernel(uint32_t* __restrict__ ubuf)
{
    __shared__ uint32_t su[N_];       // 16 KB
    __shared__ int      spart[256];
    __shared__ int      hist[256];
    __shared__ int      sChosen, sCumAbove;

    const int tid = threadIdx.x;
    const int b   = blockIdx.x;
    uint32_t* __restrict__ ub = ubuf + (size_t)b * N_;

    for (int i = 0; i < 16; ++i) {    // coalesced load
        const int n = tid + 256 * i;
        su[n] = ub[n];
    }
    __syncthreads();

    uint32_t prefix = 0;
    int kneed = K_;                    // how many still to place, 1..hist[chosen]
    for (int pass = 0; pass < 4; ++pass) {
        const int shift = 24 - 8 * pass;
        hist[tid] = 0;
        __syncthreads();
        for (int i = 0; i < 16; ++i) {
            const uint32_t v = su[tid * 16 + i];
            const bool match = (pass == 0) ? true : ((v >> (shift + 8)) == prefix);
            if (match) atomicAdd(&hist[(v >> shift) & 255], 1);
        }
        __syncthreads();
        if (tid == 0) {
            int cum = 0;
            int chosen = 0, cumAbove = 0;
            for (int d = 255; d >= 0; --d) {
                const int h = hist[d];
                if (cum + h >= kneed) { chosen = d; cumAbove = cum; break; }
                cum += h;
            }
            sChosen = chosen;
            sCumAbove = cumAbove;
        }
        __syncthreads();
        prefix = (prefix << 8) | (uint32_t)sChosen;
        kneed -= sCumAbove;
    }
    const uint32_t T = prefix;         // K-th largest key; kneed ties to keep

    // eqrank[n] = #(m < n, key==T)  (stable tie-break = lowest index first)
    int eqr[16];
    for (int i = 0; i < 16; ++i) eqr[i] = (su[tid * 16 + i] == T) ? 1 : 0;
    ts_block_exscan16(eqr, spart, tid);

    int fl[16], pos[16];
    for (int i = 0; i < 16; ++i) {
        const uint32_t v = su[tid * 16 + i];
        fl[i]  = (v > T) || (v == T && eqr[i] < kneed);
        pos[i] = fl[i];
    }
    ts_block_exscan16(pos, spart, tid); // pos = rank among kept, ascending n

    int* __restrict__ kp = (int*)ub;    // alias: overwrite keys with keeppos
    for (int i = 0; i < 16; ++i) {
        const int n = tid * 16 + i;
        kp[n] = fl[i] ? pos[i] : -1;
    }
}

// ---------------------------------------------------------------------------
// Kernel 3: tiled transpose-gather (D,N) -> (K,D) of kept tokens.
// grid = (N/32, D/32, B), block = 256. Coalesced 128B on both sides.
// ---------------------------------------------------------------------------
__global__ __launch_bounds__(256)
void ts_gather_kernel(const float* __restrict__ tokens,
                      const int* __restrict__ keeppos,
                      float* __restrict__ out)
{
    __shared__ float tile[32][33];
    __shared__ int   spos[32];

    const int tid = threadIdx.x;
    const int tx  = tid & 31;          // n-offset on load, d-offset on store
    const int ty  = tid >> 5;          // 0..7
    const int n0  = blockIdx.x * 32;
    const int d0  = blockIdx.y * 32;
    const int b   = blockIdx.z;

    const float* __restrict__ tb = tokens + (size_t)b * D_ * N_;
    for (int i = 0; i < 4; ++i) {
        const int dl = ty + 8 * i;
        tile[dl][tx] = tb[(size_t)(d0 + dl) * N_ + n0 + tx];
    }
    if (tid < 32) spos[tid] = keeppos[(size_t)b * N_ + n0 + tid];
    __syncthreads();

    for (int i = 0; i < 4; ++i) {
        const int nl = ty + 8 * i;
        const int p  = spos[nl];
        if (p >= 0) {
            out[((size_t)b * K_ + p) * D_ + d0 + tx] = tile[nl][tx];
        }
    }
}

// ---------------------------------------------------------------------------
// Launch: inputs = {tokens, norm_w, norm_b, fc_w, fc_b} (norm_b/fc_b are
// additive constants on the scores -> ordering-irrelevant, unused).
// ---------------------------------------------------------------------------
extern "C" void kernel_launch(void* const* d_in, const int* in_sizes, int n_in,
                              void* d_out, int out_size, void* d_ws, size_t ws_size,
                              hipStream_t stream)
{
    (void)in_sizes; (void)n_in; (void)out_size; (void)ws_size;
    const float* tokens = (const float*)d_in[0];
    const float* norm_w = (const float*)d_in[1];
    const float* fc_w   = (const float*)d_in[3];
    float* out = (float*)d_out;

    uint32_t* ubuf = (uint32_t*)d_ws;     // B*N u32 keys, then reused as keeppos

    dim3 gScore(N_ / 128, B_, 1);
    ts_score_kernel<<<gScore, 256, 0, stream>>>(tokens, norm_w, fc_w, ubuf);

    ts_topk_kernel<<<dim3(B_, 1, 1), 256, 0, stream>>>(ubuf);

    dim3 gGather(N_ / 32, D_ / 32, B_);
    ts_gather_kernel<<<gGather, 256, 0, stream>>>(tokens, (const int*)ubuf, out);
}